// CKGConv_66271345377495
// MI455X (gfx1250) — compile-verified
//
#include <hip/hip_runtime.h>

typedef float v2f __attribute__((ext_vector_type(2)));
typedef float v8f __attribute__((ext_vector_type(8)));

#define HID 32
#define NHEAD 8
#define CLAMP_V 5.0f

// D = A(16x4, f32) * B(4x16, f32) + C(16x16, f32), full fp32 WMMA.
__device__ __forceinline__ v8f wmma4(v2f a, v2f b, v8f c) {
    return __builtin_amdgcn_wmma_f32_16x16x4_f32(false, a, false, b, (short)0, c,
                                                 false, false);
}

// CDNA5 async copy: 16B per lane, global -> LDS, tracked by ASYNCcnt.
// inst_offset is added to BOTH the LDS and global addresses (ISA 08, §4.4),
// so one (ldsAddr, globalAddr) pair covers the whole 2KB tile with 4 issues.
__device__ __forceinline__ void async_b128(unsigned ldsAddr, unsigned long long ga,
                                           int ofs) {
    asm volatile("global_load_async_to_lds_b128 %0, %1, off offset:%2"
                 :: "v"(ldsAddr), "v"(ga), "i"(ofs) : "memory");
}
__device__ __forceinline__ void wait_async0() {
    asm volatile("s_wait_asynccnt 0x0" ::: "memory");
}

// ---------------------------------------------------------------------------
// Kernel 1: V_h = x @ WV + bV   ([N,32] @ [32,32])
// One wave per 16-row tile; K split into 8 chunks of 4; N split into 2x16.
// Input staged via async-to-LDS; output staged via LDS -> b128 stores.
// ---------------------------------------------------------------------------
__global__ void __launch_bounds__(256) vproj_kernel(
    const float* __restrict__ x, const float* __restrict__ WV,
    const float* __restrict__ bV, float* __restrict__ Vh, int nNodes)
{
    __shared__ float lds[8 * 1024];      // per wave: 512 in-stage + 512 out-stage
    const int lane = threadIdx.x & 31;
    const int wv   = threadIdx.x >> 5;
    const int hl   = lane >> 4;          // half of wave: K offset 0 or 2
    const int ln   = lane & 15;          // row (A) / col (B,D) index
    float* st  = &lds[wv * 1024];
    float* stO = st + 512;

    // B fragments of WV: chunk k rows 4k..4k+3, ntile n cols 16n..16n+15
    v2f wf[8][2];
#pragma unroll
    for (int k = 0; k < 8; ++k)
#pragma unroll
        for (int n = 0; n < 2; ++n) {
            const int r0 = 4 * k + 2 * hl;
            wf[k][n].x = WV[r0 * HID + 16 * n + ln];
            wf[k][n].y = WV[(r0 + 1) * HID + 16 * n + ln];
        }
    const float bvr0 = bV[ln], bvr1 = bV[16 + ln];

    const unsigned stLds  = (unsigned)(size_t)st + (unsigned)lane * 16u;
    const int tiles = (nNodes + 15) >> 4;
    const int wid   = blockIdx.x * (blockDim.x >> 5) + wv;
    const int nw    = gridDim.x * (blockDim.x >> 5);

    for (int t = wid; t < tiles; t += nw) {
        const int base = t << 4;
        const bool full = (base + 16 <= nNodes);
        if (full) {
            const unsigned long long ga =
                (unsigned long long)(x + (size_t)base * HID) + (unsigned)lane * 16u;
            async_b128(stLds, ga, 0);
            async_b128(stLds, ga, 512);
            async_b128(stLds, ga, 1024);
            async_b128(stLds, ga, 1536);
            wait_async0();
        } else {
            for (int e = 0; e < 16; ++e) {
                const int row = base + e;
                st[e * HID + lane] = (row < nNodes) ? x[(size_t)row * HID + lane] : 0.0f;
            }
        }
#pragma unroll
        for (int n = 0; n < 2; ++n) {
            v8f c;
            const float bias = n ? bvr1 : bvr0;
#pragma unroll
            for (int r = 0; r < 8; ++r) c[r] = bias;
#pragma unroll
            for (int k = 0; k < 8; ++k) {
                v2f a;
                a.x = st[ln * HID + 4 * k + 2 * hl];
                a.y = st[ln * HID + 4 * k + 2 * hl + 1];
                c = wmma4(a, wf[k][n], c);
            }
#pragma unroll
            for (int r = 0; r < 8; ++r)
                stO[(r + 8 * hl) * HID + 16 * n + ln] = c[r];
        }
        if (full) {
#pragma unroll
            for (int i = 0; i < 4; ++i) {
                const float4 v = *(const float4*)(stO + i * 128 + lane * 4);
                *(float4*)(Vh + (size_t)base * HID + i * 128 + lane * 4) = v;
            }
        } else {
            for (int e = 0; e < 16; ++e) {
                const int row = base + e;
                if (row < nNodes) Vh[(size_t)row * HID + lane] = stO[e * HID + lane];
            }
        }
    }
}

// ---------------------------------------------------------------------------
// Kernel 2: per-edge MLP + gather/scale/scatter.
// One wave per 16-edge tile, persistent waves, weights in registers.
// ---------------------------------------------------------------------------
__global__ void __launch_bounds__(256) edge_kernel(
    const float* __restrict__ edge_attr, const int* __restrict__ ei,
    const float* __restrict__ Vh,
    const float* __restrict__ g1, const float* __restrict__ a1,
    const float* __restrict__ W1, const float* __restrict__ b1,
    const float* __restrict__ g2, const float* __restrict__ a2,
    const float* __restrict__ W2, const float* __restrict__ b2,
    const float* __restrict__ g3, const float* __restrict__ a3,
    const float* __restrict__ Wf, const float* __restrict__ bf,
    float* __restrict__ sacc, float* __restrict__ cnt, int ne)
{
    __shared__ float lds[8 * 1152];      // per wave: 512 (E) + 512 (H) + 128 (score)
    const int lane = threadIdx.x & 31;
    const int wv   = threadIdx.x >> 5;
    const int hl   = lane >> 4;
    const int ln   = lane & 15;
    float* stE = &lds[wv * 1152];
    float* stH = stE + 512;
    float* ssc = stH + 512;

    // ---- preload weights as B fragments, affines as lane-resident scalars ----
    v2f w1f[8][2], w2f[8][2], wff[8], g1f[8], a1f[8];
#pragma unroll
    for (int k = 0; k < 8; ++k) {
        const int r0 = 4 * k + 2 * hl;
#pragma unroll
        for (int n = 0; n < 2; ++n) {
            w1f[k][n].x = W1[r0 * HID + 16 * n + ln];
            w1f[k][n].y = W1[(r0 + 1) * HID + 16 * n + ln];
            w2f[k][n].x = W2[r0 * HID + 16 * n + ln];
            w2f[k][n].y = W2[(r0 + 1) * HID + 16 * n + ln];
        }
        wff[k].x = (ln < NHEAD) ? Wf[r0 * NHEAD + ln] : 0.0f;       // pad cols 8..15
        wff[k].y = (ln < NHEAD) ? Wf[(r0 + 1) * NHEAD + ln] : 0.0f;
        g1f[k].x = g1[r0];     g1f[k].y = g1[r0 + 1];
        a1f[k].x = a1[r0];     a1f[k].y = a1[r0 + 1];
    }
    float b1r[2] = { b1[ln], b1[16 + ln] };
    float b2r[2] = { b2[ln], b2[16 + ln] };
    float g2r[2] = { g2[ln], g2[16 + ln] };
    float a2r[2] = { a2[ln], a2[16 + ln] };
    float g3r[2] = { g3[ln], g3[16 + ln] };
    float a3r[2] = { a3[ln], a3[16 + ln] };
    const float bfr = (ln < NHEAD) ? bf[ln] : 0.0f;

    const unsigned stLds = (unsigned)(size_t)stE + (unsigned)lane * 16u;
    const int tiles = (ne + 15) >> 4;
    const int wid   = blockIdx.x * (blockDim.x >> 5) + wv;
    const int nw    = gridDim.x * (blockDim.x >> 5);

    for (int t = wid; t < tiles; t += nw) {
        const int base = t << 4;
        if (t + nw < tiles)                                // warm L2 for next tile
            __builtin_prefetch(edge_attr + (size_t)(t + nw) * 512, 0, 0);

        // ---- stage raw edge_attr tile (16x32 contiguous) via async-to-LDS ----
        if (base + 16 <= ne) {
            const unsigned long long ga =
                (unsigned long long)(edge_attr + (size_t)base * HID) + (unsigned)lane * 16u;
            async_b128(stLds, ga, 0);
            async_b128(stLds, ga, 512);
            async_b128(stLds, ga, 1024);
            async_b128(stLds, ga, 1536);
            wait_async0();
        } else {
            for (int e = 0; e < 16; ++e) {
                const int row = base + e;
                stE[e * HID + lane] = (row < ne) ? edge_attr[(size_t)row * HID + lane] : 0.0f;
            }
        }

        // ---- GEMM1: h = (E*g1+a1) @ W1 + b1 ; then fold g2/a2; stage to stH ----
#pragma unroll
        for (int n = 0; n < 2; ++n) {
            v8f c;
#pragma unroll
            for (int r = 0; r < 8; ++r) c[r] = b1r[n];
#pragma unroll
            for (int k = 0; k < 8; ++k) {
                const int c0 = 4 * k + 2 * hl;
                v2f a;
                a.x = stE[ln * HID + c0]     * g1f[k].x + a1f[k].x;
                a.y = stE[ln * HID + c0 + 1] * g1f[k].y + a1f[k].y;
                c = wmma4(a, w1f[k][n], c);
            }
#pragma unroll
            for (int r = 0; r < 8; ++r)
                stH[(r + 8 * hl) * HID + 16 * n + ln] = c[r] * g2r[n] + a2r[n];
        }

        // ---- GEMM2: h2 = stH @ W2 + b2 ; e = (E + h2)*g3 + a3 -> overwrite stE ----
#pragma unroll
        for (int n = 0; n < 2; ++n) {
            v8f c;
#pragma unroll
            for (int r = 0; r < 8; ++r) c[r] = b2r[n];
#pragma unroll
            for (int k = 0; k < 8; ++k) {
                const int c0 = 4 * k + 2 * hl;
                v2f a;
                a.x = stH[ln * HID + c0];
                a.y = stH[ln * HID + c0 + 1];
                c = wmma4(a, w2f[k][n], c);
            }
#pragma unroll
            for (int r = 0; r < 8; ++r) {
                const int off = (r + 8 * hl) * HID + 16 * n + ln;
                stE[off] = (c[r] + stE[off]) * g3r[n] + a3r[n];
            }
        }

        // ---- GEMM3: score = e @ Wf(+pad) + bf ; clamp ; stage [16 x 8] ----
        {
            v8f c;
#pragma unroll
            for (int r = 0; r < 8; ++r) c[r] = bfr;
#pragma unroll
            for (int k = 0; k < 8; ++k) {
                const int c0 = 4 * k + 2 * hl;
                v2f a;
                a.x = stE[ln * HID + c0];
                a.y = stE[ln * HID + c0 + 1];
                c = wmma4(a, wff[k], c);
            }
#pragma unroll
            for (int r = 0; r < 8; ++r) {
                const float s = fminf(fmaxf(c[r], -CLAMP_V), CLAMP_V);
                if (ln < NHEAD) ssc[(r + 8 * hl) * NHEAD + ln] = s;
            }
        }

        // ---- gather V_h[src], scale by score, scatter-add to sacc[dst] ----
        int srcE = 0, dstE = 0;
        const int eIdx = base + ln;
        if (lane < 16 && eIdx < ne) {
            srcE = ei[eIdx];
            dstE = ei[ne + eIdx];
            atomicAdd(&cnt[dstE], 1.0f);
        }
        const int m = (ne - base < 16) ? (ne - base) : 16;
#pragma unroll 1
        for (int e = 0; e < m; ++e) {
            const int s = __shfl(srcE, e, 32);
            const int d = __shfl(dstE, e, 32);
            const float sc = ssc[e * NHEAD + (lane >> 2)];           // head = lane/4
            const float v  = Vh[((unsigned)s << 5) + (unsigned)lane]; // coalesced row
            atomicAdd(&sacc[((unsigned)d << 5) + (unsigned)lane], v * sc);
        }
    }
}

// ---------------------------------------------------------------------------
// Kernel 3: out = sacc / max(cnt,1) + out_bias
// ---------------------------------------------------------------------------
__global__ void __launch_bounds__(256) finalize_kernel(
    const float* __restrict__ sacc, const float* __restrict__ cnt,
    const float* __restrict__ ob, float* __restrict__ out, int total)
{
    const int i = blockIdx.x * blockDim.x + threadIdx.x;
    if (i < total)
        out[i] = sacc[i] / fmaxf(cnt[i >> 5], 1.0f) + ob[i & 31];
}

extern "C" void kernel_launch(void* const* d_in, const int* in_sizes, int n_in,
                              void* d_out, int out_size, void* d_ws, size_t ws_size,
                              hipStream_t stream)
{
    const float* x         = (const float*)d_in[0];
    const float* edge_attr = (const float*)d_in[1];
    const int*   ei        = (const int*)  d_in[2];
    const float* WV = (const float*)d_in[3];
    const float* bV = (const float*)d_in[4];
    const float* g1 = (const float*)d_in[5];
    const float* a1 = (const float*)d_in[6];
    const float* W1 = (const float*)d_in[7];
    const float* b1 = (const float*)d_in[8];
    const float* g2 = (const float*)d_in[9];
    const float* a2 = (const float*)d_in[10];
    const float* W2 = (const float*)d_in[11];
    const float* b2 = (const float*)d_in[12];
    const float* g3 = (const float*)d_in[13];
    const float* a3 = (const float*)d_in[14];
    const float* Wf = (const float*)d_in[15];
    const float* bf = (const float*)d_in[16];
    const float* ob = (const float*)d_in[17];

    const int nNodes = in_sizes[0] / HID;
    const int ne     = in_sizes[1] / HID;

    float* Vh   = (float*)d_ws;
    float* sacc = Vh + (size_t)nNodes * HID;
    float* cnt  = sacc + (size_t)nNodes * HID;

    hipMemsetAsync(sacc, 0, ((size_t)nNodes * HID + nNodes) * sizeof(float), stream);

    const int tilesA = (nNodes + 15) / 16;
    int gridA = (tilesA + 7) / 8;
    if (gridA < 1) gridA = 1;
    vproj_kernel<<<gridA, 256, 0, stream>>>(x, WV, bV, Vh, nNodes);

    const int tilesB = (ne + 15) / 16;
    int gridB = (tilesB + 7) / 8;
    if (gridB > 2048) gridB = 2048;
    if (gridB < 1) gridB = 1;
    edge_kernel<<<gridB, 256, 0, stream>>>(edge_attr, ei, Vh, g1, a1, W1, b1,
                                           g2, a2, W2, b2, g3, a3, Wf, bf,
                                           sacc, cnt, ne);

    const int total = nNodes * HID;
    finalize_kernel<<<(total + 255) / 256, 256, 0, stream>>>(sacc, cnt, ob,
                                                             (float*)d_out, total);
}